// OpfGNN_56435870270044
// MI455X (gfx1250) — compile-verified
//
#include <hip/hip_runtime.h>
#include <hip/hip_bf16.h>

typedef __attribute__((ext_vector_type(2))) float v2f;
typedef __attribute__((ext_vector_type(8))) float v8f;

#define F_IN  128
#define F_HID 64
#define F_OUT 2

// ---------------------------------------------------------------------------
// Utility: zero a float buffer (accumulators must be re-zeroed every call).
// ---------------------------------------------------------------------------
__global__ __launch_bounds__(256) void gcn_zero_f32(float* __restrict__ p, long long count) {
    long long i = (long long)blockIdx.x * blockDim.x + threadIdx.x;
    if (i < count) p[i] = 0.0f;
}

// deg[i] = 1.0 (self loop contribution)
__global__ __launch_bounds__(256) void gcn_deg_init(float* __restrict__ deg, int n) {
    int i = blockIdx.x * blockDim.x + threadIdx.x;
    if (i < n) deg[i] = 1.0f;
}

// deg[dst[e]] += 1 over real edges
__global__ __launch_bounds__(256) void gcn_deg_count(const int* __restrict__ dst,
                                                     float* __restrict__ deg, int e) {
    int t = blockIdx.x * blockDim.x + threadIdx.x;
    if (t < e) atomicAdd(&deg[dst[t]], 1.0f);
}

// deg -> rsqrt(deg) in place (deg >= 1 always due to self loops)
__global__ __launch_bounds__(256) void gcn_dinv(float* __restrict__ deg, int n) {
    int i = blockIdx.x * blockDim.x + threadIdx.x;
    if (i < n) deg[i] = rsqrtf(deg[i]);
}

// ---------------------------------------------------------------------------
// Layer-1 GEMM: h = x @ W1, full fp32 via V_WMMA_F32_16X16X4_F32.
// Block = 128 threads = 4 waves; wave w computes the 16x16 tile at
// rows [16*blockIdx.x, +16), cols [16*w, +16). K loop: 128/4 = 32 WMMAs.
//
// A (16x4 f32):  lane L: M = L&15; VGPRs hold K = {0,1} (L<16) / {2,3} (L>=16)
// B (4x16 f32):  lane L: N = L&15; VGPRs hold K = {0,1} (L<16) / {2,3} (L>=16)
// C/D (16x16):   VGPR j: M = j + 8*(L>=16), N = L&15
// ---------------------------------------------------------------------------
__global__ __launch_bounds__(128) void gcn_gemm1_wmma(const float* __restrict__ x,
                                                      const float* __restrict__ W1,
                                                      float* __restrict__ h) {
    const int lane  = threadIdx.x & 31;
    const int wave  = threadIdx.x >> 5;          // 0..3 -> column tile
    const int row0  = blockIdx.x * 16;
    const int col0  = wave * 16;
    const int m     = lane & 15;                 // M (for A) / N (for B/D)
    const int khalf = (lane >> 4) * 2;           // 0 or 2

    const float* xrow = x + (size_t)(row0 + m) * F_IN;
    v8f acc = {};

#pragma unroll 4
    for (int k = 0; k < F_IN; k += 4) {
        v2f a, b;
        a.x = xrow[k + khalf + 0];
        a.y = xrow[k + khalf + 1];
        b.x = W1[(size_t)(k + khalf + 0) * F_HID + col0 + m];
        b.y = W1[(size_t)(k + khalf + 1) * F_HID + col0 + m];
        acc = __builtin_amdgcn_wmma_f32_16x16x4_f32(
            /*neg_a=*/false, a, /*neg_b=*/false, b,
            /*c_mod=*/(short)0, acc, /*reuse_a=*/false, /*reuse_b=*/false);
    }

    const int mbase = (lane >> 4) * 8;
    float* hp = h + (size_t)row0 * F_HID + col0 + m;
#pragma unroll
    for (int j = 0; j < 8; ++j)
        hp[(size_t)(mbase + j) * F_HID] = acc[j];
}

// Scalar tail for n % 16 != 0 (not launched for n = 100000).
__global__ __launch_bounds__(64) void gcn_gemm1_tail(const float* __restrict__ x,
                                                     const float* __restrict__ W1,
                                                     float* __restrict__ h,
                                                     int start, int n) {
    int idx  = blockIdx.x * blockDim.x + threadIdx.x;
    int node = start + idx / F_HID;
    int col  = idx % F_HID;
    if (node >= n) return;
    const float* xr = x + (size_t)node * F_IN;
    float acc = 0.0f;
    for (int k = 0; k < F_IN; ++k)
        acc = fmaf(xr[k], W1[(size_t)k * F_HID + col], acc);
    h[(size_t)node * F_HID + col] = acc;
}

// ---------------------------------------------------------------------------
// Layer-1 aggregation: agg[dst] += h[src] * (dinv[src]*dinv[dst]).
// One thread per (edge, 4-feature chunk): 16 threads/edge, float4 gathers.
// Edge ids >= n_edges are the self loops (src = dst = e - n_edges).
// h and agg (25.6 MB each) are L2-resident on MI455X (192 MB L2).
// ---------------------------------------------------------------------------
__global__ __launch_bounds__(256) void gcn_agg1(const float* __restrict__ h,
                                                const float* __restrict__ dinv,
                                                const int* __restrict__ src,
                                                const int* __restrict__ dst,
                                                float* __restrict__ agg,
                                                int n_edges, int n_nodes) {
    long long t = (long long)blockIdx.x * blockDim.x + threadIdx.x;
    long long total = ((long long)n_edges + n_nodes) * (F_HID / 4);
    if (t >= total) return;
    int e = (int)(t >> 4);
    int c = ((int)t & 15) * 4;
    int s, d;
    if (e < n_edges) { s = src[e]; d = dst[e]; }
    else             { s = d = e - n_edges; }
    float nrm = dinv[s] * dinv[d];
    const float4 v = *(const float4*)(h + (size_t)s * F_HID + c);
    float* ap = agg + (size_t)d * F_HID + c;
    atomicAdd(ap + 0, v.x * nrm);
    atomicAdd(ap + 1, v.y * nrm);
    atomicAdd(ap + 2, v.z * nrm);
    atomicAdd(ap + 3, v.w * nrm);
}

// hr = relu(agg + b1), written over the h buffer (h no longer needed).
__global__ __launch_bounds__(256) void gcn_bias_relu(const float* __restrict__ agg,
                                                     const float* __restrict__ b1,
                                                     float* __restrict__ hr,
                                                     long long count) {
    long long t = (long long)blockIdx.x * blockDim.x + threadIdx.x;
    if (t >= count) return;
    hr[t] = fmaxf(agg[t] + b1[(int)(t & (F_HID - 1))], 0.0f);
}

// Layer-2 GEMM: h2 = hr @ W2 ([N x 64] @ [64 x 2]); N dim of 2 makes WMMA
// wasteful — memory-bound scalar dot is the right shape here.
__global__ __launch_bounds__(256) void gcn_gemm2(const float* __restrict__ hr,
                                                 const float* __restrict__ W2,
                                                 float* __restrict__ h2, int n) {
    int i = blockIdx.x * blockDim.x + threadIdx.x;
    if (i >= n) return;
    const float* r = hr + (size_t)i * F_HID;
    float o0 = 0.0f, o1 = 0.0f;
#pragma unroll 8
    for (int k = 0; k < F_HID; ++k) {
        float v = r[k];
        o0 = fmaf(v, W2[k * F_OUT + 0], o0);
        o1 = fmaf(v, W2[k * F_OUT + 1], o1);
    }
    h2[(size_t)i * F_OUT + 0] = o0;
    h2[(size_t)i * F_OUT + 1] = o1;
}

// Layer-2 aggregation: one thread per edge, 2 atomic adds.
__global__ __launch_bounds__(256) void gcn_agg2(const float* __restrict__ h2,
                                                const float* __restrict__ dinv,
                                                const int* __restrict__ src,
                                                const int* __restrict__ dst,
                                                float* __restrict__ agg2,
                                                int n_edges, int n_nodes) {
    long long t = (long long)blockIdx.x * blockDim.x + threadIdx.x;
    if (t >= (long long)n_edges + n_nodes) return;
    int e = (int)t;
    int s, d;
    if (e < n_edges) { s = src[e]; d = dst[e]; }
    else             { s = d = e - n_edges; }
    float nrm = dinv[s] * dinv[d];
    atomicAdd(&agg2[(size_t)d * F_OUT + 0], h2[(size_t)s * F_OUT + 0] * nrm);
    atomicAdd(&agg2[(size_t)d * F_OUT + 1], h2[(size_t)s * F_OUT + 1] * nrm);
}

// out = (agg2 + b2) * (x[:,0] == 1.0), flattened [N*2].
__global__ __launch_bounds__(256) void gcn_finalize(const float* __restrict__ agg2,
                                                    const float* __restrict__ b2,
                                                    const float* __restrict__ x,
                                                    float* __restrict__ out, int n) {
    int i = blockIdx.x * blockDim.x + threadIdx.x;
    if (i >= n) return;
    float mask = (x[(size_t)i * F_IN] == 1.0f) ? 1.0f : 0.0f;
    out[(size_t)i * F_OUT + 0] = (agg2[(size_t)i * F_OUT + 0] + b2[0]) * mask;
    out[(size_t)i * F_OUT + 1] = (agg2[(size_t)i * F_OUT + 1] + b2[1]) * mask;
}

// ---------------------------------------------------------------------------
extern "C" void kernel_launch(void* const* d_in, const int* in_sizes, int n_in,
                              void* d_out, int out_size, void* d_ws, size_t ws_size,
                              hipStream_t stream) {
    const float* x  = (const float*)d_in[0];
    const int*   ei = (const int*)d_in[1];     // edge_index [2, E] (int32 per harness)
    const float* W1 = (const float*)d_in[2];
    const float* b1 = (const float*)d_in[3];
    const float* W2 = (const float*)d_in[4];
    const float* b2 = (const float*)d_in[5];
    float* out = (float*)d_out;

    const int n = in_sizes[0] / F_IN;          // 100000
    const int E = in_sizes[1] / 2;             // 3200000
    const int* src = ei;
    const int* dst = ei + E;

    // Workspace layout (floats), dinv block padded to a multiple of 256:
    float* ws = (float*)d_ws;
    size_t o = 0;
    float* dinv = ws + o;  o += ((size_t)n + 255) & ~(size_t)255;
    float* h    = ws + o;  o += (size_t)n * F_HID;   // h, later reused for relu(h)
    float* agg  = ws + o;  o += (size_t)n * F_HID;
    float* h2   = ws + o;  o += (size_t)n * F_OUT;
    float* agg2 = ws + o;  o += (size_t)n * F_OUT;
    (void)ws_size; (void)n_in; (void)out_size;

    const int B = 256;

    // 1) degrees -> dinv (in place)
    gcn_deg_init<<<(n + B - 1) / B, B, 0, stream>>>(dinv, n);
    gcn_deg_count<<<(E + B - 1) / B, B, 0, stream>>>(dst, dinv, E);
    gcn_dinv<<<(n + B - 1) / B, B, 0, stream>>>(dinv, n);

    // 2) zero accumulators (ws is poisoned; also needed on every replay)
    const long long c1 = (long long)n * F_HID;
    const long long c2 = (long long)n * F_OUT;
    gcn_zero_f32<<<(unsigned)((c1 + B - 1) / B), B, 0, stream>>>(agg, c1);
    gcn_zero_f32<<<(unsigned)((c2 + B - 1) / B), B, 0, stream>>>(agg2, c2);

    // 3) h = x @ W1 (WMMA fp32)
    if (n >= 16)
        gcn_gemm1_wmma<<<n / 16, 128, 0, stream>>>(x, W1, h);
    if (n % 16) {
        int start = (n / 16) * 16;
        int tail  = (n - start) * F_HID;
        gcn_gemm1_tail<<<(tail + 63) / 64, 64, 0, stream>>>(x, W1, h, start, n);
    }

    // 4) agg = scatter_add(h[src] * norm)
    const long long t1 = ((long long)E + n) * (F_HID / 4);
    gcn_agg1<<<(unsigned)((t1 + B - 1) / B), B, 0, stream>>>(h, dinv, src, dst, agg, E, n);

    // 5) hr = relu(agg + b1)  (overwrites h buffer)
    gcn_bias_relu<<<(unsigned)((c1 + B - 1) / B), B, 0, stream>>>(agg, b1, h, c1);

    // 6) h2 = hr @ W2
    gcn_gemm2<<<(n + B - 1) / B, B, 0, stream>>>(h, W2, h2, n);

    // 7) agg2 = scatter_add(h2[src] * norm)
    const long long t2 = (long long)E + n;
    gcn_agg2<<<(unsigned)((t2 + B - 1) / B), B, 0, stream>>>(h2, dinv, src, dst, agg2, E, n);

    // 8) out = (agg2 + b2) * mask
    gcn_finalize<<<(n + B - 1) / B, B, 0, stream>>>(agg2, b2, x, out, n);
}